// Base_Filter_35648228556847
// MI455X (gfx1250) — compile-verified
//
#include <hip/hip_runtime.h>
#include <stdint.h>

// Problem constants (from reference): x is (256,256,32,32) fp32, k = ix*32+iy is contiguous.
#define A_DIM   256
#define K_DIM   1024
#define AO      250        // valid output size (256 - 7 + 1)
#define KS      7
#define TAPS    49
#define TROWS   8          // output rows per block (one per wave)
#define TCOLS   16         // output cols per thread
#define INROWS  (TROWS + KS - 1)   // 14
#define INCOLS  (TCOLS + KS - 1)   // 22
#define TK      32         // channels per block (one per lane, wave32)

typedef unsigned int uint32x4 __attribute__((ext_vector_type(4)));
typedef int          int32x8  __attribute__((ext_vector_type(8)));
typedef int          int32x4  __attribute__((ext_vector_type(4)));

__global__ __launch_bounds__(256)
void depthwise7x7_kernel(const float* __restrict__ x,
                         const float* __restrict__ w_p,
                         const float* __restrict__ b_p,
                         const float* __restrict__ v,
                         const float* __restrict__ g,
                         float* __restrict__ out) {
    __shared__ float tile[INROWS * INCOLS * TK];   // [row][col][k], k fastest: 38.5 KB

    const int tid  = threadIdx.x;
    const int lane = tid & 31;
    const int wave = tid >> 5;

    const int kg = blockIdx.x;            // channel group 0..31
    const int b0 = blockIdx.y * TCOLS;    // output col base 0..240
    const int a0 = blockIdx.z * TROWS;    // output row base 0..248
    const int k  = (kg << 5) + lane;      // this thread's channel

    // ------------------------------------------------------------------
    // Stage input tile x[a0..a0+13][b0..b0+21][kg*32..kg*32+31] into LDS.
    // ------------------------------------------------------------------
#if defined(__gfx1250__) && __has_builtin(__builtin_amdgcn_tensor_load_to_lds)
    if (wave == 0) {   // exactly one full wave issues the TDM op (EXEC all ones)
        const uint64_t gaddr =
            (uint64_t)(uintptr_t)(x + (((uint64_t)(a0 * A_DIM + b0)) << 10) + (kg << 5));
        const uint32_t ldsa = (uint32_t)(uintptr_t)(&tile[0]);   // LDS byte offset

        // D# group 0: count=1, lds_addr, global_addr[56:0], type=2 ("image")
        uint32x4 g0;
        g0.x = 1u;
        g0.y = ldsa;
        g0.z = (uint32_t)gaddr;
        g0.w = ((uint32_t)((gaddr >> 32) & 0x01FFFFFFull)) | 0x80000000u;

        // D# group 1: data_size=4B; tensor dims = REMAINING extents so that
        // edge-tile halo reads past the image zero-fill instead of faulting.
        const uint32_t td0 = (uint32_t)K_DIM;           // k extent
        const uint32_t td1 = (uint32_t)(A_DIM - b0);    // remaining b extent
        const uint64_t s0  = (uint64_t)K_DIM;           // stride between cols (elements)
        const uint64_t s1  = (uint64_t)A_DIM * K_DIM;   // stride between rows (elements)
        int32x8 g1;
        g1[0] = (int)(2u << 16);                                      // data_size=2 -> 4 bytes
        g1[1] = (int)((td0 & 0xFFFFu) << 16);                         // tensor_dim0 lo16
        g1[2] = (int)((td0 >> 16) | ((td1 & 0xFFFFu) << 16));         // td0 hi16 | td1 lo16
        g1[3] = (int)((td1 >> 16) | ((uint32_t)TK << 16));            // td1 hi16 | tile_dim0=32
        g1[4] = (int)(((uint32_t)INCOLS) | (((uint32_t)INROWS) << 16)); // tile_dim1=22, tile_dim2=14
        g1[5] = (int)(uint32_t)(s0 & 0xFFFFFFFFull);                  // dim0 stride lo32
        g1[6] = (int)((uint32_t)(s0 >> 32) | ((uint32_t)(s1 & 0xFFFFull) << 16));
        g1[7] = (int)(uint32_t)(s1 >> 16);                            // dim1 stride hi32

        // D# group 2: tensor_dim2 = remaining a extent; no dim3, no iterate
        int32x4 g2;
        g2.x = (int)(uint32_t)(A_DIM - a0);
        g2.y = 0; g2.z = 0; g2.w = 0;
        int32x4 g3; g3.x = 0; g3.y = 0; g3.z = 0; g3.w = 0;

        // amdgpu-toolchain (clang-23) 6-arg form: extra v8i32 operand before cpol.
        int32x8 gx;
        gx[0] = 0; gx[1] = 0; gx[2] = 0; gx[3] = 0;
        gx[4] = 0; gx[5] = 0; gx[6] = 0; gx[7] = 0;

        __builtin_amdgcn_tensor_load_to_lds(g0, g1, g2, g3, gx, 0);
        __builtin_amdgcn_s_wait_tensorcnt(0);
    }
    __syncthreads();
#else
    // Fallback: plain coalesced staging (lane == channel -> 128B per wave access)
    for (int idx = tid; idx < INROWS * INCOLS * TK; idx += 256) {
        const int kk  = idx & 31;
        const int rc  = idx >> 5;
        const int col = rc % INCOLS;
        const int row = rc / INCOLS;
        const int a = a0 + row, b = b0 + col;
        float val = 0.0f;
        if (a < A_DIM && b < A_DIM)
            val = x[(((uint64_t)(a * A_DIM + b)) << 10) + (kg << 5) + kk];
        tile[idx] = val;
    }
    __syncthreads();
#endif

    // ------------------------------------------------------------------
    // Per-channel weight-norm: w_eff = g * v / ||v||_F (49 taps in regs).
    // Affine fold: conv((1+w_p)x + b_p) = (1+w_p)*conv(x) + b_p*sum(w_eff)
    // ------------------------------------------------------------------
    const float* vk = v + (uint64_t)k * TAPS;
    float w[TAPS];
    float ss = 0.0f;
#pragma unroll
    for (int t = 0; t < TAPS; ++t) { w[t] = vk[t]; ss = fmaf(w[t], w[t], ss); }
    const float wn = g[k] / sqrtf(ss);
    float wsum = 0.0f;
#pragma unroll
    for (int t = 0; t < TAPS; ++t) { w[t] *= wn; wsum += w[t]; }
    const float sc = 1.0f + w_p[k];
    const float bs = b_p[k] * wsum;

    // ------------------------------------------------------------------
    // 7x7 stencil: 784 FMAs vs 154 ds_load_b32 per thread (row reuse).
    // ------------------------------------------------------------------
    float acc[TCOLS];
#pragma unroll
    for (int n = 0; n < TCOLS; ++n) acc[n] = 0.0f;

#pragma unroll
    for (int i = 0; i < KS; ++i) {
        const float* r = &tile[((wave + i) * INCOLS) * TK + lane];
        float rv[INCOLS];
#pragma unroll
        for (int c = 0; c < INCOLS; ++c) rv[c] = r[c * TK];   // conflict-free: banks k-contig
#pragma unroll
        for (int j = 0; j < KS; ++j) {
            const float wij = w[i * KS + j];
#pragma unroll
            for (int n = 0; n < TCOLS; ++n)
                acc[n] = fmaf(wij, rv[n + j], acc[n]);
        }
    }

    // ------------------------------------------------------------------
    // Affine + leaky-decay relu + store (coalesced over k).
    // ------------------------------------------------------------------
    const int arow = a0 + wave;
    if (arow < AO) {
        float* orow = out + (((uint64_t)(arow * AO + b0)) << 10) + k;
        const int nb = (AO - b0 < TCOLS) ? (AO - b0) : TCOLS;
        for (int n = 0; n < nb; ++n) {
            const float rr = fmaf(sc, acc[n], bs);
            orow[(uint64_t)n << 10] = (rr > 0.0f) ? 0.9f * rr : 0.01f * rr;
        }
    }
}

extern "C" void kernel_launch(void* const* d_in, const int* in_sizes, int n_in,
                              void* d_out, int out_size, void* d_ws, size_t ws_size,
                              hipStream_t stream) {
    const float* x   = (const float*)d_in[0];
    const float* w_p = (const float*)d_in[1];
    const float* b_p = (const float*)d_in[2];
    const float* v   = (const float*)d_in[3];
    const float* g   = (const float*)d_in[4];
    float* out = (float*)d_out;

    dim3 grid(K_DIM / TK,                       // 32 channel groups
              (AO + TCOLS - 1) / TCOLS,        // 16 col tiles
              (AO + TROWS - 1) / TROWS);       // 32 row tiles
    depthwise7x7_kernel<<<grid, dim3(256), 0, stream>>>(x, w_p, b_p, v, g, out);
}